// RTM3DLoss_67491116089496
// MI455X (gfx1250) — compile-verified
//
#include <hip/hip_runtime.h>

typedef __attribute__((ext_vector_type(2)))  float    v2f;
typedef __attribute__((ext_vector_type(8)))  float    v8f;

#define BB 8
#define CC 3
#define HH 96
#define WW 320
#define NBOX 4096
#define MAXR 31
#define KW 63                       // window width = 2*MAXR+1
#define HEAT_ELEMS (BB*CC*HH*WW)    // 737280

__device__ __forceinline__ float gauss_radius(float bw, float bh) {
  const float mo = 0.7f;            // MIN_OVERLAP
  float h = ceilf(bh), w = ceilf(bw);
  float b1 = h + w;
  float c1 = w * h * (1.f - mo) / (1.f + mo);
  float r1 = 0.5f * (b1 + sqrtf(b1 * b1 - 4.f * c1));
  float b2 = 2.f * (h + w);
  float c2 = (1.f - mo) * w * h;
  float r2 = 0.5f * (b2 + sqrtf(b2 * b2 - 16.f * c2));
  float b3 = -2.f * mo * (h + w);
  float c3 = (mo - 1.f) * w * h;
  float r3 = 0.5f * (b3 + sqrtf(b3 * b3 - 16.f * mo * c3));
  float r = fminf(r1, fminf(r2, r3));
  return fminf(fmaxf(r, 0.f), (float)MAXR);
}

// min |wx - 31| over wx in [16t, 16t+15]  (tile fully outside radius iff > r)
__device__ __forceinline__ int tile_min_absd(int t) {
  int lo = 16 * t - MAXR;          // dx at tile start
  int hi = 16 * t - 16;            // dx at tile end
  if (lo <= 0 && hi >= 0) return 0;
  int alo = lo < 0 ? -lo : lo;
  int ahi = hi < 0 ? -hi : hi;
  return alo < ahi ? alo : ahi;
}

__global__ void zero_heat(float* __restrict__ out) {
  int i = blockIdx.x * blockDim.x + threadIdx.x;
  if (i < HEAT_ELEMS) out[i] = 0.f;
}

__global__ void proj_offset(const float* __restrict__ bboxes, float* __restrict__ out) {
  int n = blockIdx.x * blockDim.x + threadIdx.x;
  if (n >= NBOX) return;
  float x1 = bboxes[4*n+0], y1 = bboxes[4*n+1];
  float x2 = bboxes[4*n+2], y2 = bboxes[4*n+3];
  float cx = 0.5f * (x1 + x2), cy = 0.5f * (y1 + y2);
  int px = (int)cx, py = (int)cy;                 // trunc-toward-zero == astype(int32)
  float* proj = out + HEAT_ELEMS;
  float* offs = out + HEAT_ELEMS + 2 * NBOX;
  proj[2*n+0] = (float)px;
  proj[2*n+1] = (float)py;
  offs[2*n+0] = cx - (float)px;
  offs[2*n+1] = cy - (float)py;
}

// One block (128 thr = 4 waves) per box. Wave w owns 16 window rows
// [16w, 16w+16); loops 4 tile-columns. Each tile: rank-1 outer product
// gy (16x1) x gx (1x16) via the K=0 slice of V_WMMA_F32_16X16X4_F32, then
// per-lane scatter-max of its 8 accumulator elements. Tiles fully outside
// the box radius are skipped wave-uniformly (EXEC stays all-ones at WMMA).
__global__ void __launch_bounds__(128) splat(const float* __restrict__ bboxes,
                                             const int*   __restrict__ classes,
                                             const int*   __restrict__ img_id,
                                             const int*   __restrict__ noise,
                                             float*       __restrict__ heat) {
  const int n     = blockIdx.x;
  const int lane  = threadIdx.x & 31;
  const int tileM = threadIdx.x >> 5;   // 0..3 (wave id)

  const float x1 = bboxes[4*n+0], y1 = bboxes[4*n+1];
  const float x2 = bboxes[4*n+2], y2 = bboxes[4*n+3];
  const float cx = 0.5f * (x1 + x2), cy = 0.5f * (y1 + y2);

  const float r      = gauss_radius(x2 - x1, y2 - y1);
  const float sigma  = (2.f * r + 1.f) / 6.f;
  const float inv2s2 = 1.f / (2.f * sigma * sigma);
  const int   ri     = (int)r;          // |d| <= r  <=>  |d| <= floor(r) for int d

  // Wave-uniform early out: this wave's 16 rows all outside the radius.
  // (tileM==1 contains dy==0, never skipped -> noise center tap always handled.)
  if (tile_min_absd(tileM) > ri) return;

  const int  cls     = classes[n];
  const int  img     = img_id[n];
  const bool isNoise = noise[n] != 0;
  float* base = heat + ((size_t)img * CC + cls) * (size_t)(HH * WW);

  // gy for this wave's rows; K=0 slice lives in lanes 0-15 of A's first VGPR.
  float gy = 0.f;
  {
    int   wy = tileM * 16 + lane;
    float dy = (float)(wy - MAXR);
    if (lane < 16 && wy < KW && fabsf(dy) <= r)
      gy = expf(-dy * dy * inv2s2);
  }

  // Per-lane row targets, hoisted out of the tile-column loop:
  // C/D layout: VGPR j -> M = j + (lane>=16 ? 8 : 0), N = lane & 15.
  const int rowBase = (lane >> 4) * 8;
  int  ty8[8];
  bool yok8[8];
  #pragma unroll
  for (int j = 0; j < 8; ++j) {
    int wy  = tileM * 16 + rowBase + j;
    int ty  = (int)(cy + (float)(wy - MAXR));   // trunc toward zero
    ty8[j]  = ty;
    yok8[j] = (wy < KW) && (ty >= 0) && (ty < HH);
  }

  const int  colN = lane & 15;

  #pragma unroll
  for (int tileN = 0; tileN < 4; ++tileN) {
    // Wave-uniform skip of fully-outside tile columns (saves WMMA + atomics).
    if (tile_min_absd(tileN) > ri) continue;

    float gx = 0.f;
    {
      int   wx = tileN * 16 + lane;
      float dx = (float)(wx - MAXR);
      if (lane < 16 && wx < KW && fabsf(dx) <= r)
        gx = expf(-dx * dx * inv2s2);
    }

    v2f a = {gy, 0.f};                      // only K=0 slice non-zero
    v2f b = {gx, 0.f};
    v8f c = {};
    c = __builtin_amdgcn_wmma_f32_16x16x4_f32(false, a, false, b,
                                              (short)0, c, false, false);

    const int   wx  = tileN * 16 + colN;
    const int   tx  = (int)(cx + (float)(wx - MAXR));
    const bool  xok = (wx < KW) && (tx >= 0) && (tx < WW);

    #pragma unroll
    for (int j = 0; j < 8; ++j) {
      float val = c[j];
      if (tileN == 1 && isNoise && wx == MAXR &&
          (tileM * 16 + rowBase + j) == MAXR)
        val = 0.9999f;                      // noise center tap override
      if (!xok || !yok8[j]) continue;
      if (val <= 0.f) continue;             // max(0, heat) is a no-op
      // heat >= 0 everywhere: unsigned bit-pattern order == float order
      atomicMax((unsigned int*)&base[ty8[j] * WW + tx], __float_as_uint(val));
    }
  }
}

extern "C" void kernel_launch(void* const* d_in, const int* in_sizes, int n_in,
                              void* d_out, int out_size, void* d_ws, size_t ws_size,
                              hipStream_t stream) {
  // d_in order: main_kf_logits(f32, shape only), bboxes(f32), classes(i32),
  //             img_id(i32), noise_mask(int), max_radius(i32 scalar, == 31)
  const float* bboxes  = (const float*)d_in[1];
  const int*   classes = (const int*)  d_in[2];
  const int*   img     = (const int*)  d_in[3];
  const int*   noise   = (const int*)  d_in[4];
  float* out = (float*)d_out;

  zero_heat  <<<(HEAT_ELEMS + 255) / 256, 256, 0, stream>>>(out);
  proj_offset<<<(NBOX + 255) / 256,       256, 0, stream>>>(bboxes, out);
  splat      <<<NBOX,                     128, 0, stream>>>(bboxes, classes, img, noise, out);
}